// HE_FM_87196426043503
// MI455X (gfx1250) — compile-verified
//
#include <hip/hip_runtime.h>

// FM inference: out[r] = w[u] + w[i] + b + dot(V[u], V[i]), EMBED_DIM = 64.
// Memory-bound gather (~8.7 MB, ~0.4 us at 23.3 TB/s); dot products done as the
// diagonal of a 16x16 WMMA tile product (V_WMMA_F32_16X16X4_F32, K accumulated
// 64 = 16 chunks of 4) since matrix FLOPs are free at this arithmetic intensity.

typedef __attribute__((ext_vector_type(2))) float v2f;
typedef __attribute__((ext_vector_type(8))) float v8f;

#define FM_USER_NUM 1000000
#define FM_EMBED_DIM 64
#define FM_BATCH 16384

__global__ __launch_bounds__(256) void fm_wmma_diag_kernel(
    const long long* __restrict__ INPUT,  // (BATCH, 2) int64
    const float* __restrict__ w,          // (USER_NUM + ITEM_NUM)
    const float* __restrict__ b,          // (1)
    const float* __restrict__ V,          // (USER_NUM + ITEM_NUM, 64)
    float* __restrict__ out)              // (BATCH)
{
    const int lane = threadIdx.x & 31;          // wave32
    const int wave = threadIdx.x >> 5;          // 8 waves / block
    const int tile = blockIdx.x * 8 + wave;     // 16 batch rows per wave-tile
    const int rowt = lane & 15;                 // row within tile (A: M, B: N)
    const int half = lane >> 4;                 // lane half selects k pair

    const int r = tile * 16 + rowt;             // global batch row for this lane

    // Per-lane indices (each row's indices replicated across the two halves).
    const long long u64 = INPUT[2 * (long long)r + 0];
    const long long i64 = INPUT[2 * (long long)r + 1];
    const int u = (int)u64;
    const int i = (int)i64 + FM_USER_NUM;

    // A-matrix layout (16x4 f32): lane (half,rowt), vgpr v holds
    // A[rowt][k0 + 2*half + v]  -> contiguous float2 at offset k0 + 2*half.
    // B-matrix (4x16 f32) uses the same k -> (vgpr, half) map with N = rowt,
    // giving the identical float2 pattern on the Vi rows.
    const float* baseU = V + ((long long)u * FM_EMBED_DIM + 2 * half);
    const float* baseI = V + ((long long)i * FM_EMBED_DIM + 2 * half);

    // Hoist all 32 global_load_b64 so the wave keeps them in flight (latency
    // hiding is the whole game for this random-gather workload).
    v2f av[16];
    v2f bv[16];
#pragma unroll
    for (int k = 0; k < 16; ++k) {
        av[k] = *(const v2f*)(baseU + 4 * k);
        bv[k] = *(const v2f*)(baseI + 4 * k);
    }

    // Accumulate D = sum_k0 A(:,k0:k0+4) * B(k0:k0+4,:) ; D[m][m] = dot(Vu_m, Vi_m).
    v8f c = {};
#pragma unroll
    for (int k = 0; k < 16; ++k) {
        c = __builtin_amdgcn_wmma_f32_16x16x4_f32(
            /*neg_a=*/false, av[k],
            /*neg_b=*/false, bv[k],
            /*c_mod=*/(short)0, c,
            /*reuse_a=*/false, /*reuse_b=*/false);
    }

    // Diagonal of the 16x16 f32 C/D layout:
    //   m in 0..7  -> lane m      , component m
    //   m in 8..15 -> lane m + 16 , component m - 8
    // => lanes 0-7 and 24-31 hold diag in component (lane & 7).
    const int ci = lane & 7;
    float d = c[0];
    d = (ci == 1) ? c[1] : d;
    d = (ci == 2) ? c[2] : d;
    d = (ci == 3) ? c[3] : d;
    d = (ci == 4) ? c[4] : d;
    d = (ci == 5) ? c[5] : d;
    d = (ci == 6) ? c[6] : d;
    d = (ci == 7) ? c[7] : d;

    const bool valid = (lane < 8) || (lane >= 24);
    if (valid) {
        const float lin = w[u] + w[i] + b[0];
        out[r] = lin + d;   // lanes 0-7 -> rows 0-7, lanes 24-31 -> rows 8-15
    }
}

extern "C" void kernel_launch(void* const* d_in, const int* in_sizes, int n_in,
                              void* d_out, int out_size, void* d_ws, size_t ws_size,
                              hipStream_t stream) {
    (void)in_sizes; (void)n_in; (void)d_ws; (void)ws_size; (void)out_size;

    const long long* INPUT = (const long long*)d_in[0];  // int64 (BATCH,2)
    const float*     w     = (const float*)d_in[1];
    const float*     b     = (const float*)d_in[2];
    const float*     V     = (const float*)d_in[3];
    float*           out   = (float*)d_out;

    // 16384 rows / 16 rows-per-wave = 1024 waves = 128 blocks * 8 waves.
    dim3 grid(FM_BATCH / 16 / 8);
    dim3 block(256);
    fm_wmma_diag_kernel<<<grid, block, 0, stream>>>(INPUT, w, b, V, out);
}